// DiffusionCoordinateInitializer_91207925498228
// MI455X (gfx1250) — compile-verified
//
#include <hip/hip_runtime.h>

typedef float v2f __attribute__((ext_vector_type(2)));
typedef float v8f __attribute__((ext_vector_type(8)));

namespace {

constexpr int kWavesPerBlock = 4;
constexpr int kTilesPerWave  = 2;                              // 2 x 16 rows
constexpr int kRowsPerWave   = kTilesPerWave * 16;             // 32
constexpr int kRowsPerBlock  = kWavesPerBlock * kRowsPerWave;  // 128

// Run the reference's unrolled diffusion recurrence for one output row and store.
__device__ __forceinline__ void diffuse_store(int row, int rows, int T,
                                              float c0, float c1, float c2,
                                              const float* __restrict__ noise,
                                              float* __restrict__ out)
{
    if (row >= rows) return;
    const size_t o = (size_t)row * 3;
    float x0 = noise[o + 0];
    float x1 = noise[o + 1];
    float x2 = noise[o + 2];
    for (int t = T - 1; t >= 0; --t) {
        // Match Python: alpha = (t+1)/T in double, rounded to f32 at use.
        const double ad = (double)(t + 1) / (double)T;
        const float  al = (float)ad;
        const float  be = (float)(1.0 - ad);
        x0 = al * x0 + be * c0;
        x1 = al * x1 + be * c1;
        x2 = al * x2 + be * c2;
    }
    out[o + 0] = x0;
    out[o + 1] = x1;
    out[o + 2] = x2;
}

// coords = latent @ W^T + b  via V_WMMA_F32_16X16X4_F32, then diffusion epilogue.
// A matrix = W chunk (16x4, rows 0..2 valid, rest zeroed by mask -> no divergence).
// B matrix = 16 latent rows x 4 K (4x16, K x N).
// D[n, m]: n = coord (acc VGPRs 0..2), m = latent row (lanes 0..15).
__global__ __launch_bounds__(kWavesPerBlock * 32)
void diffusion_coords_wmma(const float* __restrict__ latent,
                           const float* __restrict__ W,
                           const float* __restrict__ bias,
                           const float* __restrict__ noise,
                           const int*   __restrict__ stepsPtr,
                           float* __restrict__ out,
                           int rows, int D)
{
    const int lane   = threadIdx.x & 31;
    const int waveId = threadIdx.x >> 5;
    const int laneN  = lane & 15;          // A: W row index / B: latent row in tile
    const int khalf  = (lane >> 4) << 1;   // 0 for lanes 0-15, 2 for lanes 16-31

    const int rowBase = ((int)blockIdx.x * kWavesPerBlock + waveId) * kRowsPerWave;
    if (rowBase >= rows) return;

    // A tile source: W rows 0..2; lanes with laneN>=3 read row 0 and multiply by 0,
    // keeping EXEC uniform through the hot loop (WMMA requires EXEC all-ones).
    const float wmask = (laneN < 3) ? 1.0f : 0.0f;
    const int   wrow  = (laneN < 3) ? laneN : 0;
    const float* wp = W + (size_t)wrow * D + khalf;

    // B tile sources: two 16-row tiles of latent (row clamped for tail safety).
    const int r0 = min(rowBase +      laneN, rows - 1);
    const int r1 = min(rowBase + 16 + laneN, rows - 1);
    const float* lp0 = latent + (size_t)r0 * D + khalf;
    const float* lp1 = latent + (size_t)r1 * D + khalf;

    v8f acc0 = {0.f, 0.f, 0.f, 0.f, 0.f, 0.f, 0.f, 0.f};
    v8f acc1 = {0.f, 0.f, 0.f, 0.f, 0.f, 0.f, 0.f, 0.f};

#pragma unroll 4
    for (int k = 0; k < D; k += 4) {
        // A: lane holds W[wrow, k+khalf .. +1] (8B aligned b64 load), masked.
        v2f a = *(const v2f*)(wp + k);
        a = a * wmask;
        // B: lane holds latent[row, k+khalf .. +1]; one A load feeds two WMMAs.
        const v2f b0 = *(const v2f*)(lp0 + k);
        const v2f b1 = *(const v2f*)(lp1 + k);
        acc0 = __builtin_amdgcn_wmma_f32_16x16x4_f32(false, a, false, b0,
                                                     (short)0, acc0, false, false);
        acc1 = __builtin_amdgcn_wmma_f32_16x16x4_f32(false, a, false, b1,
                                                     (short)0, acc1, false, false);
    }

    const int   T   = *stepsPtr;
    const float bb0 = bias[0];
    const float bb1 = bias[1];
    const float bb2 = bias[2];

    // D layout: lanes 0-15 hold n=0..7 in acc[0..7] for m=lane; we need n=0..2.
    if (lane < 16) {
        diffuse_store(rowBase + lane, rows, T,
                      acc0[0] + bb0, acc0[1] + bb1, acc0[2] + bb2, noise, out);
        diffuse_store(rowBase + 16 + lane, rows, T,
                      acc1[0] + bb0, acc1[1] + bb1, acc1[2] + bb2, noise, out);
    }
}

} // namespace

extern "C" void kernel_launch(void* const* d_in, const int* in_sizes, int n_in,
                              void* d_out, int out_size, void* d_ws, size_t ws_size,
                              hipStream_t stream)
{
    const float* latent = (const float*)d_in[0];   // [B, S, D] f32
    const float* W      = (const float*)d_in[1];   // [3, D]    f32
    const float* bias   = (const float*)d_in[2];   // [3]       f32
    const float* noise  = (const float*)d_in[3];   // [B, S, 3] f32
    const int*   steps  = (const int*)d_in[4];     // scalar int

    const int rows = in_sizes[3] / 3;              // B*S = 32768
    const int D    = in_sizes[0] / rows;           // 2048

    dim3 block(kWavesPerBlock * 32);
    dim3 grid((rows + kRowsPerBlock - 1) / kRowsPerBlock);
    hipLaunchKernelGGL(diffusion_coords_wmma, grid, block, 0, stream,
                       latent, W, bias, noise, steps, (float*)d_out, rows, D);
}